// MultiHeadAttention_70428873719870
// MI455X (gfx1250) — compile-verified
//
#include <hip/hip_runtime.h>
#include <math.h>

typedef __attribute__((ext_vector_type(2))) float v2f;
typedef __attribute__((ext_vector_type(8))) float v8f;

typedef int v4i __attribute__((vector_size(16)));
typedef __attribute__((address_space(1))) v4i glob_v4i;
typedef __attribute__((address_space(3))) v4i lds_v4i;

#define kB 8
#define kS 1024
#define kD 768
#define kH 12
#define kHD 64
#define k3D 2304

__device__ __forceinline__ v8f wmma4(v2f a, v2f b, v8f c) {
  // D = A(16x4) * B(4x16) + C, fp32 in/out
  return __builtin_amdgcn_wmma_f32_16x16x4_f32(false, a, false, b, (short)0, c,
                                               false, false);
}

// 16-byte global->LDS copy on the CDNA5 async-to-LDS path (ASYNCcnt).
__device__ __forceinline__ void copy16_async(void* lds, const void* g) {
#if __has_builtin(__builtin_amdgcn_global_load_async_to_lds_b128)
  __builtin_amdgcn_global_load_async_to_lds_b128((glob_v4i*)(void*)g,
                                                 (lds_v4i*)lds, 0, 0);
#else
  *(float4*)lds = *(const float4*)g;
#endif
}

#define WAIT_ASYNC(n) asm volatile("s_wait_asynccnt " #n ::: "memory")

// ---------------------------------------------------------------------------
// Shared GEMM body: 8 waves share one 64-col W panel staged in LDS.
// LDS layout is k-pair interleaved (Wp[kpair][col][2]) so every B fragment is
// one contiguous ds_load_b64; kpair stride 160 floats => the two lane halves
// (kpair and kpair+1) hit disjoint 32-bank halves (160 mod 64 == 32).
// ---------------------------------------------------------------------------
#define WP_STRIDE 160

// ---------------------------------------------------------------------------
// Kernel 1: QKV projection.  C[8192,2304] = X[8192,768] @ W[768,2304],
// scattered into Q/K/V laid out [B,H,S,HD].
// ---------------------------------------------------------------------------
__global__ __launch_bounds__(256) void mha_qkv_gemm(
    const float* __restrict__ x, const float* __restrict__ w,
    float* __restrict__ Q, float* __restrict__ K, float* __restrict__ V) {
  __shared__ float Wp[2][16][WP_STRIDE];  // 16 kpairs (32 k-rows) x 64 cols x2

  const int lane = threadIdx.x & 31;
  const int wave = threadIdx.x >> 5;
  const int l16 = lane & 15;
  const int hlf = lane >> 4;
  const int rowTile = blockIdx.x * 8 + wave;  // 0..511
  const int row = rowTile * 16 + l16;         // A-matrix row for this lane
  const int colBase = blockIdx.y * 64;        // 0..2240 step 64

  const int sr = threadIdx.x >> 4;        // staging k-row 0..15 (and +16)
  const int sc = (threadIdx.x & 15) * 4;  // staging col 0..60
  auto stage_w = [&](int kb, int buf) {
    const float4 t0 =
        *(const float4*)(w + (size_t)(kb * 32 + sr) * k3D + colBase + sc);
    const float4 t1 =
        *(const float4*)(w + (size_t)(kb * 32 + sr + 16) * k3D + colBase + sc);
    const int par = sr & 1;
    float* p0 = &Wp[buf][sr >> 1][0];
    float* p1 = &Wp[buf][(sr + 16) >> 1][0];
    p0[2 * (sc + 0) + par] = t0.x;
    p0[2 * (sc + 1) + par] = t0.y;
    p0[2 * (sc + 2) + par] = t0.z;
    p0[2 * (sc + 3) + par] = t0.w;
    p1[2 * (sc + 0) + par] = t1.x;
    p1[2 * (sc + 1) + par] = t1.y;
    p1[2 * (sc + 2) + par] = t1.z;
    p1[2 * (sc + 3) + par] = t1.w;
  };

  v8f acc0 = {}, acc1 = {}, acc2 = {}, acc3 = {};
  const float* xrow = x + (size_t)row * kD;

  stage_w(0, 0);
  for (int kb = 0; kb < kD / 32; ++kb) {
    __syncthreads();  // WAR on next buffer + visibility of current buffer
    if (kb + 1 < kD / 32) stage_w(kb + 1, (kb + 1) & 1);
    const int cb = kb & 1;
#pragma unroll
    for (int kc = 0; kc < 8; ++kc) {
      const int k = kb * 32 + kc * 4;
      v2f a = *(const v2f*)(xrow + k + 2 * hlf);  // A[m][k..k+1]
      const float* wp = &Wp[cb][2 * kc + hlf][0];
      v2f b0 = *(const v2f*)(wp + 2 * (0 + l16));
      v2f b1 = *(const v2f*)(wp + 2 * (16 + l16));
      v2f b2 = *(const v2f*)(wp + 2 * (32 + l16));
      v2f b3 = *(const v2f*)(wp + 2 * (48 + l16));
      acc0 = wmma4(a, b0, acc0);
      acc1 = wmma4(a, b1, acc1);
      acc2 = wmma4(a, b2, acc2);
      acc3 = wmma4(a, b3, acc3);
    }
  }

  const int which = blockIdx.y / kH;  // 0=Q 1=K 2=V
  const int h = blockIdx.y % kH;
  float* dst = (which == 0) ? Q : (which == 1) ? K : V;
  const int batch = (rowTile * 16) / kS;
  const int sBase = (rowTile * 16) % kS;

#pragma unroll
  for (int r = 0; r < 8; ++r) {
    const int srow = sBase + r + 8 * hlf;
    const size_t base = (((size_t)(batch * kH + h)) * kS + srow) * kHD;
    dst[base + 0 + l16] = acc0[r];
    dst[base + 16 + l16] = acc1[r];
    dst[base + 32 + l16] = acc2[r];
    dst[base + 48 + l16] = acc3[r];
  }
}

// ---------------------------------------------------------------------------
// Kernel 2: causal flash attention for one (b,h) slice.
// grid = (S/128, B*H); block = 256 (8 waves); wave handles 16 queries.
// K/V tiles are double-buffered in LDS via CDNA5 async-to-LDS copies.
// ---------------------------------------------------------------------------
__global__ __launch_bounds__(256) void mha_attention(
    const float* __restrict__ Q, const float* __restrict__ K,
    const float* __restrict__ V, float* __restrict__ out) {
  __shared__ float Kt[2][16][68];   // K^T-fragment reads want stride 68
  __shared__ float Vt[2][16][72];   // row-uniform reads want stride 72
  __shared__ float Pst[8][16][18];  // per-wave P-tile staging (C->A relayout)

  const int lane = threadIdx.x & 31;
  const int wave = threadIdx.x >> 5;
  const int l16 = lane & 15;
  const int hlf = lane >> 4;
  const int bh = blockIdx.y;
  const int h = bh % kH;
  const int batch = bh / kH;
  const int Qb = blockIdx.x * 128;
  const int qbase = Qb + wave * 16;
  const int qblk = qbase >> 4;

  const float* Qbh = Q + (size_t)bh * kS * kHD;
  const float* Kbh = K + (size_t)bh * kS * kHD;
  const float* Vbh = V + (size_t)bh * kS * kHD;

  const int sr = threadIdx.x >> 4;        // staging row 0..15
  const int sc = (threadIdx.x & 15) * 4;  // staging col 0..60
  auto stage_kv = [&](int j, int buf) {
    copy16_async(&Kt[buf][sr][sc], Kbh + (size_t)(j * 16 + sr) * kHD + sc);
    copy16_async(&Vt[buf][sr][sc], Vbh + (size_t)(j * 16 + sr) * kHD + sc);
  };

  // Preload Q tile as 16 A-fragments covering HD=64 (16 chunks of K=4).
  v2f qreg[16];
#pragma unroll
  for (int kc = 0; kc < 16; ++kc)
    qreg[kc] =
        *(const v2f*)(Qbh + (size_t)(qbase + l16) * kHD + 4 * kc + 2 * hlf);

  v8f o0 = {}, o1 = {}, o2 = {}, o3 = {};
  float mrow[8], lrow[8];
#pragma unroll
  for (int r = 0; r < 8; ++r) {
    mrow[r] = -__builtin_inff();
    lrow[r] = 0.0f;
  }

  const int jmax = (Qb >> 4) + 7;  // uniform over the block
  stage_kv(0, 0);
  for (int j = 0; j <= jmax; ++j) {
    __syncthreads();  // all waves done reading buffer (j+1)&1
    if (j < jmax) {
      stage_kv(j + 1, (j + 1) & 1);
      WAIT_ASYNC(2);  // tile j (2 older copies) has landed
    } else {
      WAIT_ASYNC(0);
    }
    __syncthreads();  // all waves' copies of tile j visible
    const int cb = j & 1;
    if (j > qblk) continue;  // fully-masked tile (wave-uniform skip)

    // S = Q @ K^T for this 16x16 tile: 16 WMMA f32 16x16x4 steps.
    v8f s = {};
#pragma unroll
    for (int kc = 0; kc < 16; ++kc) {
      v2f bK = *(const v2f*)&Kt[cb][l16][4 * kc + 2 * hlf];
      s = wmma4(qreg[kc], bK, s);
    }

    // Scale (x8), causal mask, online softmax update, stage P into LDS.
    const int kg = j * 16 + l16;
#pragma unroll
    for (int r = 0; r < 8; ++r) {
      const int qg = qbase + r + 8 * hlf;
      float val = s[r] * 8.0f;
      val = (kg > qg) ? -__builtin_inff() : val;
      float tm = val;
      tm = fmaxf(tm, __shfl_xor(tm, 1, 16));
      tm = fmaxf(tm, __shfl_xor(tm, 2, 16));
      tm = fmaxf(tm, __shfl_xor(tm, 4, 16));
      tm = fmaxf(tm, __shfl_xor(tm, 8, 16));
      const float mnew = fmaxf(mrow[r], tm);
      const float p = __expf(val - mnew);
      float ps = p;
      ps += __shfl_xor(ps, 1, 16);
      ps += __shfl_xor(ps, 2, 16);
      ps += __shfl_xor(ps, 4, 16);
      ps += __shfl_xor(ps, 8, 16);
      const float alpha = __expf(mrow[r] - mnew);
      lrow[r] = lrow[r] * alpha + ps;
      mrow[r] = mnew;
      o0[r] *= alpha;
      o1[r] *= alpha;
      o2[r] *= alpha;
      o3[r] *= alpha;
      Pst[wave][r + 8 * hlf][l16] = p;
    }
    asm volatile("s_wait_dscnt 0" ::: "memory");

    // O += P @ V: 4 chunks of K=4 over the 16 keys, 4 N-tiles of HD.
#pragma unroll
    for (int kc = 0; kc < 4; ++kc) {
      v2f pa = *(const v2f*)&Pst[wave][l16][4 * kc + 2 * hlf];
      const int vr = 4 * kc + 2 * hlf;
      v2f b0 = {Vt[cb][vr][0 + l16], Vt[cb][vr + 1][0 + l16]};
      v2f b1 = {Vt[cb][vr][16 + l16], Vt[cb][vr + 1][16 + l16]};
      v2f b2 = {Vt[cb][vr][32 + l16], Vt[cb][vr + 1][32 + l16]};
      v2f b3 = {Vt[cb][vr][48 + l16], Vt[cb][vr + 1][48 + l16]};
      o0 = wmma4(pa, b0, o0);
      o1 = wmma4(pa, b1, o1);
      o2 = wmma4(pa, b2, o2);
      o3 = wmma4(pa, b3, o3);
    }
  }

#pragma unroll
  for (int r = 0; r < 8; ++r) {
    const int qg = qbase + r + 8 * hlf;
    const float inv = 1.0f / lrow[r];
    const size_t base = ((size_t)batch * kS + qg) * kD + h * kHD;
    out[base + 0 + l16] = o0[r] * inv;
    out[base + 16 + l16] = o1[r] * inv;
    out[base + 32 + l16] = o2[r] * inv;
    out[base + 48 + l16] = o3[r] * inv;
  }
}

// ---------------------------------------------------------------------------
// Kernel 3: output projection.  out[8192,768] = A @ W_proj + b_proj.
// Same pair-interleaved LDS pipeline as kernel 1.
// ---------------------------------------------------------------------------
__global__ __launch_bounds__(256) void mha_proj_gemm(
    const float* __restrict__ a, const float* __restrict__ w,
    const float* __restrict__ bias, float* __restrict__ out) {
  __shared__ float Wp[2][16][WP_STRIDE];

  const int lane = threadIdx.x & 31;
  const int wave = threadIdx.x >> 5;
  const int l16 = lane & 15;
  const int hlf = lane >> 4;
  const int rowTile = blockIdx.x * 8 + wave;
  const int row = rowTile * 16 + l16;
  const int colBase = blockIdx.y * 64;

  const int sr = threadIdx.x >> 4;
  const int sc = (threadIdx.x & 15) * 4;
  auto stage_w = [&](int kb, int buf) {
    const float4 t0 =
        *(const float4*)(w + (size_t)(kb * 32 + sr) * kD + colBase + sc);
    const float4 t1 =
        *(const float4*)(w + (size_t)(kb * 32 + sr + 16) * kD + colBase + sc);
    const int par = sr & 1;
    float* p0 = &Wp[buf][sr >> 1][0];
    float* p1 = &Wp[buf][(sr + 16) >> 1][0];
    p0[2 * (sc + 0) + par] = t0.x;
    p0[2 * (sc + 1) + par] = t0.y;
    p0[2 * (sc + 2) + par] = t0.z;
    p0[2 * (sc + 3) + par] = t0.w;
    p1[2 * (sc + 0) + par] = t1.x;
    p1[2 * (sc + 1) + par] = t1.y;
    p1[2 * (sc + 2) + par] = t1.z;
    p1[2 * (sc + 3) + par] = t1.w;
  };

  v8f acc0 = {}, acc1 = {}, acc2 = {}, acc3 = {};
  const float* arow = a + (size_t)row * kD;

  stage_w(0, 0);
  for (int kb = 0; kb < kD / 32; ++kb) {
    __syncthreads();
    if (kb + 1 < kD / 32) stage_w(kb + 1, (kb + 1) & 1);
    const int cb = kb & 1;
#pragma unroll
    for (int kc = 0; kc < 8; ++kc) {
      const int k = kb * 32 + kc * 4;
      v2f av = *(const v2f*)(arow + k + 2 * hlf);
      const float* wp = &Wp[cb][2 * kc + hlf][0];
      v2f b0 = *(const v2f*)(wp + 2 * (0 + l16));
      v2f b1 = *(const v2f*)(wp + 2 * (16 + l16));
      v2f b2 = *(const v2f*)(wp + 2 * (32 + l16));
      v2f b3 = *(const v2f*)(wp + 2 * (48 + l16));
      acc0 = wmma4(av, b0, acc0);
      acc1 = wmma4(av, b1, acc1);
      acc2 = wmma4(av, b2, acc2);
      acc3 = wmma4(av, b3, acc3);
    }
  }

  const float bi0 = bias[colBase + 0 + l16];
  const float bi1 = bias[colBase + 16 + l16];
  const float bi2 = bias[colBase + 32 + l16];
  const float bi3 = bias[colBase + 48 + l16];
#pragma unroll
  for (int r = 0; r < 8; ++r) {
    const int grow = rowTile * 16 + r + 8 * hlf;
    const size_t base = (size_t)grow * kD + colBase;
    out[base + 0 + l16] = acc0[r] + bi0;
    out[base + 16 + l16] = acc1[r] + bi1;
    out[base + 32 + l16] = acc2[r] + bi2;
    out[base + 48 + l16] = acc3[r] + bi3;
  }
}

extern "C" void kernel_launch(void* const* d_in, const int* in_sizes, int n_in,
                              void* d_out, int out_size, void* d_ws,
                              size_t ws_size, hipStream_t stream) {
  (void)in_sizes;
  (void)n_in;
  (void)out_size;
  (void)ws_size;
  const float* x = (const float*)d_in[0];       // [8,1024,768]
  const float* w_qkv = (const float*)d_in[1];   // [768,2304]
  const float* w_proj = (const float*)d_in[2];  // [768,768]
  const float* b_proj = (const float*)d_in[3];  // [768]
  float* out = (float*)d_out;                   // [8,1024,768]

  const size_t bhsd = (size_t)kB * kH * kS * kHD;  // 6291456 elems
  float* Q = (float*)d_ws;
  float* K = Q + bhsd;
  float* V = K + bhsd;
  float* attn = V + bhsd;  // [8192,768]; total ws use = 96 MB

  mha_qkv_gemm<<<dim3(64, 36), 256, 0, stream>>>(x, w_qkv, Q, K, V);
  mha_attention<<<dim3(kS / 128, kB * kH), 256, 0, stream>>>(Q, K, V, attn);
  mha_proj_gemm<<<dim3(64, 12), 256, 0, stream>>>(attn, w_proj, b_proj, out);
}